// StructuralGNN_58548994179072
// MI455X (gfx1250) — compile-verified
//
#include <hip/hip_runtime.h>

typedef float v2f __attribute__((ext_vector_type(2)));
typedef float v8f __attribute__((ext_vector_type(8)));

#define N_NODES 100000
#define N_EDGES 3200000

// ---------------------------------------------------------------------------
// Degree / normalization
// ---------------------------------------------------------------------------
__global__ void init_deg_kernel(float* __restrict__ deg, int n) {
    int i = blockIdx.x * blockDim.x + threadIdx.x;
    if (i < n) deg[i] = 1.0f;   // self-loop contributes 1
}

__global__ void deg_accum_kernel(const int* __restrict__ edst, float* __restrict__ deg,
                                 int nedges) {
    int e = blockIdx.x * blockDim.x + threadIdx.x;
    if (e < nedges) atomicAdd(&deg[edst[e]], 1.0f);
}

__global__ void finalize_dis_kernel(float* __restrict__ degdis, int n) {
    int i = blockIdx.x * blockDim.x + threadIdx.x;
    if (i < n) {
        float d = degdis[i];
        degdis[i] = (d > 0.0f) ? rsqrtf(d) : 0.0f;   // deg >= 1 always, kept for parity
    }
}

// ---------------------------------------------------------------------------
// Layer-1 dense: h = x @ W1   (K = 3, plain VALU)
// ---------------------------------------------------------------------------
__global__ void gemm_k3_kernel(const float* __restrict__ X, const float* __restrict__ W,
                               float* __restrict__ H, int n) {
    int t = blockIdx.x * blockDim.x + threadIdx.x;
    if (t >= n * 128) return;
    int i = t >> 7;          // node
    int f = t & 127;         // feature
    const float* xr = X + i * 3;
    float acc = fmaf(xr[0], W[f], fmaf(xr[1], W[128 + f], xr[2] * W[256 + f]));
    H[t] = acc;
}

// ---------------------------------------------------------------------------
// WMMA fp32 GEMM: H[n x OUT] = X[n x 128] @ W[128 x OUT]
// One wave per 16-row M-tile, covering all OUT/16 N-tiles.
// Uses V_WMMA_F32_16X16X4_F32 (ISA 7.12.2 32-bit A/B/C layouts).
// ---------------------------------------------------------------------------
template <int OUT>
__global__ void gemm_wmma_f32_kernel(const float* __restrict__ X,
                                     const float* __restrict__ W,
                                     float* __restrict__ H, int nrows) {
    const int lane  = threadIdx.x & 31;
    const int wave  = threadIdx.x >> 5;
    const int mtile = blockIdx.x * (blockDim.x >> 5) + wave;
    const int ntiles_m = nrows >> 4;          // nrows multiple of 16 (100000 = 6250*16)
    if (mtile >= ntiles_m) return;

    const int mrow = (mtile << 4) + (lane & 15);   // A-row this lane feeds
    const int koff = (lane >> 4) << 1;             // lanes 0-15: K=0,1 ; lanes 16-31: K=2,3
    constexpr int NT = OUT / 16;

    v8f acc[NT];
#pragma unroll
    for (int t = 0; t < NT; ++t) acc[t] = (v8f){0.f,0.f,0.f,0.f,0.f,0.f,0.f,0.f};

    const float* xrow = X + (long long)mrow * 128;

    for (int kk = 0; kk < 128; kk += 4) {
        v2f a;
        a.x = xrow[kk + koff + 0];
        a.y = xrow[kk + koff + 1];
#pragma unroll
        for (int t = 0; t < NT; ++t) {
            const int col = (t << 4) + (lane & 15);
            v2f b;
            b.x = W[(kk + koff + 0) * OUT + col];
            b.y = W[(kk + koff + 1) * OUT + col];
            acc[t] = __builtin_amdgcn_wmma_f32_16x16x4_f32(
                /*neg_a=*/false, a, /*neg_b=*/false, b,
                /*c_mod=*/(short)0, acc[t], /*reuse_a=*/false, /*reuse_b=*/false);
        }
    }

    // C/D layout: VGPR r -> rows (r, r+8) split across lane halves
    const int rbase = (mtile << 4) + ((lane >> 4) << 3);
    const int cbase = lane & 15;
#pragma unroll
    for (int t = 0; t < NT; ++t) {
#pragma unroll
        for (int r = 0; r < 8; ++r) {
            H[(long long)(rbase + r) * OUT + (t << 4) + cbase] = acc[t][r];
        }
    }
}

// ---------------------------------------------------------------------------
// Self-loop init: agg[i][f] = dis[i]^2 * h[i][f]   (also zero-initializes agg)
// ---------------------------------------------------------------------------
template <int F>
__global__ void selfloop_init_kernel(const float* __restrict__ dis,
                                     const float* __restrict__ h,
                                     float* __restrict__ agg, int n) {
    int t = blockIdx.x * blockDim.x + threadIdx.x;
    if (t >= n * F) return;
    int i = t / F;
    float d = dis[i];
    agg[t] = d * d * h[t];
}

// ---------------------------------------------------------------------------
// Edge scatter: one wave per edge; each lane carries F/32 contiguous floats.
// float4/float2 gathers (GLOBAL_LOAD_B128/B64) + global_atomic_add_f32.
// ---------------------------------------------------------------------------
template <int F>
__global__ void edge_scatter_kernel(const int* __restrict__ esrc,
                                    const int* __restrict__ edst,
                                    const float* __restrict__ dis,
                                    const float* __restrict__ h,
                                    float* __restrict__ agg, int nedges) {
    long long tid = (long long)blockIdx.x * blockDim.x + threadIdx.x;
    long long e = tid >> 5;
    if (e >= nedges) return;
    const int lane = (int)(tid & 31);
    const int s = esrc[e];
    const int d = edst[e];
    const float nrm = dis[s] * dis[d];

    if (F == 128) {
        const float4 v = *(const float4*)(h + (long long)s * 128 + lane * 4);
        float* dp = agg + (long long)d * 128 + lane * 4;
        atomicAdd(dp + 0, nrm * v.x);
        atomicAdd(dp + 1, nrm * v.y);
        atomicAdd(dp + 2, nrm * v.z);
        atomicAdd(dp + 3, nrm * v.w);
    } else {  // F == 64
        const float2 v = *(const float2*)(h + (long long)s * 64 + lane * 2);
        float* dp = agg + (long long)d * 64 + lane * 2;
        atomicAdd(dp + 0, nrm * v.x);
        atomicAdd(dp + 1, nrm * v.y);
    }
}

// ---------------------------------------------------------------------------
// Bias (+ optional ReLU), in place
// ---------------------------------------------------------------------------
template <int F, bool RELU>
__global__ void bias_act_kernel(float* __restrict__ a, const float* __restrict__ b, int n) {
    int t = blockIdx.x * blockDim.x + threadIdx.x;
    if (t >= n * F) return;
    float v = a[t] + b[t % F];
    a[t] = RELU ? fmaxf(v, 0.0f) : v;
}

// ---------------------------------------------------------------------------
// Launcher
// ---------------------------------------------------------------------------
extern "C" void kernel_launch(void* const* d_in, const int* in_sizes, int n_in,
                              void* d_out, int out_size, void* d_ws, size_t ws_size,
                              hipStream_t stream) {
    const float* x  = (const float*)d_in[0];
    const int*   ei = (const int*)d_in[1];       // [2, E] : row 0 = src, row 1 = dst
    const float* W1 = (const float*)d_in[2];
    const float* b1 = (const float*)d_in[3];
    const float* W2 = (const float*)d_in[4];
    const float* b2 = (const float*)d_in[5];
    const float* W3 = (const float*)d_in[6];
    const float* b3 = (const float*)d_in[7];
    float* out = (float*)d_out;

    const int n = N_NODES;
    const int E = N_EDGES;
    const int* esrc = ei;
    const int* edst = ei + E;

    // Workspace layout (fp32): dis[N] | bufA[N*128] | bufB[N*128]
    char* ws = (char*)d_ws;
    float* dis  = (float*)ws;                                   // 400,000 B (16B aligned)
    float* bufA = (float*)(ws + (size_t)n * sizeof(float));     // h buffers
    float* bufB = bufA + (size_t)n * 128;

    const int TPB = 256;
    const int gN    = (n + TPB - 1) / TPB;
    const int gE    = (E + TPB - 1) / TPB;
    const int gNF   = (n * 128 + TPB - 1) / TPB;
    const int gNF64 = (n * 64 + TPB - 1) / TPB;
    const int gEdgeWave = (int)(((long long)E * 32 + TPB - 1) / TPB);
    const int gGemm = ((n / 16) + (TPB / 32) - 1) / (TPB / 32);

    // normalization
    init_deg_kernel<<<gN, TPB, 0, stream>>>(dis, n);
    deg_accum_kernel<<<gE, TPB, 0, stream>>>(edst, dis, E);
    finalize_dis_kernel<<<gN, TPB, 0, stream>>>(dis, n);

    // ----- layer 1: h1 = x@W1 ; agg ; +b1 ; relu -----
    gemm_k3_kernel<<<gNF, TPB, 0, stream>>>(x, W1, bufA, n);
    selfloop_init_kernel<128><<<gNF, TPB, 0, stream>>>(dis, bufA, bufB, n);
    edge_scatter_kernel<128><<<gEdgeWave, TPB, 0, stream>>>(esrc, edst, dis, bufA, bufB, E);
    bias_act_kernel<128, true><<<gNF, TPB, 0, stream>>>(bufB, b1, n);

    // ----- layer 2: h2 = out1@W2 (WMMA f32) ; agg ; +b2 ; relu -----
    gemm_wmma_f32_kernel<128><<<gGemm, TPB, 0, stream>>>(bufB, W2, bufA, n);
    selfloop_init_kernel<128><<<gNF, TPB, 0, stream>>>(dis, bufA, bufB, n);
    edge_scatter_kernel<128><<<gEdgeWave, TPB, 0, stream>>>(esrc, edst, dis, bufA, bufB, E);
    bias_act_kernel<128, true><<<gNF, TPB, 0, stream>>>(bufB, b2, n);

    // ----- layer 3: h3 = out2@W3 (WMMA f32, OUT=64) ; agg into d_out ; +b3 -----
    gemm_wmma_f32_kernel<64><<<gGemm, TPB, 0, stream>>>(bufB, W3, bufA, n);
    selfloop_init_kernel<64><<<gNF64, TPB, 0, stream>>>(dis, bufA, out, n);
    edge_scatter_kernel<64><<<gEdgeWave, TPB, 0, stream>>>(esrc, edst, dis, bufA, out, E);
    bias_act_kernel<64, false><<<gNF64, TPB, 0, stream>>>(out, b3, n);
}